// SOM_58540404244877
// MI455X (gfx1250) — compile-verified
//
#include <hip/hip_runtime.h>

typedef __attribute__((ext_vector_type(2))) float v2f;
typedef __attribute__((ext_vector_type(8))) float v8f;

static constexpr int kN = 1024;        // N_MAX
static constexpr int kD = 64;          // feature dim
static constexpr int kB = 131072;      // batch rows
static constexpr int kRowsPerBlock = 64;
static constexpr int kXsStride = 68;   // 64 + 4 pad floats: float4-aligned, bank-conflict-free

// ---------------------------------------------------------------------------
// kernel 0a: zero the high-branch accumulators (h_cnt | h_sum, contiguous)
// ---------------------------------------------------------------------------
__global__ void som_zero(float* __restrict__ p, int n) {
  int i = blockIdx.x * blockDim.x + threadIdx.x;
  if (i < n) p[i] = 0.0f;
}

// ---------------------------------------------------------------------------
// kernel 0b: per-node params  np4[n] = { rsum, |w|^2, rsum/64, node_control }
// ---------------------------------------------------------------------------
__global__ __launch_bounds__(64) void som_nodeparams(
    const float* __restrict__ W, const float* __restrict__ rel,
    const float* __restrict__ nc, float4* __restrict__ np4) {
  const int n = blockIdx.x, d = threadIdx.x;
  __shared__ float s0[64], s1[64];
  float w = W[n * kD + d];
  s0[d] = rel[n * kD + d];
  s1[d] = w * w;
  __syncthreads();
  for (int s = 32; s > 0; s >>= 1) {
    if (d < s) { s0[d] += s0[d + s]; s1[d] += s1[d + s]; }
    __syncthreads();
  }
  if (d == 0) {
    float rs = s0[0];
    np4[n] = make_float4(rs, s1[0], rs * (1.0f / 64.0f), nc[n]);
  }
}

// ---------------------------------------------------------------------------
// kernel 1: fp32 WMMA GEMM (x @ W^T) + activation + per-row argmax +
//           high-row segment sums.
//   grid: kB/64 blocks x 256 threads (8 waves: 4 row-tiles x 2 node-slices)
// ---------------------------------------------------------------------------
__global__ __launch_bounds__(256) void som_gemm_assign(
    const float* __restrict__ x, const float* __restrict__ W,
    const float4* __restrict__ np4,
    float* __restrict__ h_cnt, float* __restrict__ h_sum) {
  __shared__ float xs[kRowsPerBlock * kXsStride];
  __shared__ float xn[kRowsPerBlock];
  __shared__ unsigned long long smax[kRowsPerBlock];

  const int tid = threadIdx.x;
  const long rowBase = (long)blockIdx.x * kRowsPerBlock;

  // stage 64x64 f32 x-tile into padded LDS (float4 copies)
  {
    const float4* xg = (const float4*)(x + rowBase * kD);
    float4* xs4 = (float4*)xs;
    for (int i = tid; i < kRowsPerBlock * (kD / 4); i += 256) {
      int r = i >> 4, c = i & 15;
      xs4[r * (kXsStride / 4) + c] = xg[i];
    }
  }
  if (tid < kRowsPerBlock) { xn[tid] = 0.0f; smax[tid] = 0ull; }
  __syncthreads();

  // per-row |x|^2 : 4 threads/row, rotated banks via padded stride
  {
    int r = tid >> 2, q = tid & 3;
    float s = 0.0f;
#pragma unroll
    for (int j = 0; j < 16; ++j) {
      float v = xs[r * kXsStride + q * 16 + j];
      s += v * v;
    }
    atomicAdd(&xn[r], s);
  }
  __syncthreads();

  const int wave = tid >> 5;  // 0..7
  const int lane = tid & 31;
  const int rt = wave >> 1;   // row tile 0..3 -> rows rt*16..rt*16+15
  const int ct = wave & 1;    // node sub-slice within iteration
  const int l16 = lane & 15;
  const int hi = lane >> 4;   // half-wave select
  const int koff = hi * 2;    // ISA A/B fragment K offset for lanes 16..31

  // A fragments: loop-invariant, held in registers for the whole node loop.
  v2f afrag[16];
  {
    const float* xrow = xs + (rt * 16 + l16) * kXsStride;
#pragma unroll
    for (int kk = 0; kk < 16; ++kk) {
      const float2 t = *(const float2*)(xrow + kk * 4 + koff);
      afrag[kk].x = t.x; afrag[kk].y = t.y;
    }
  }
  // |x|^2 for this lane's 8 C-rows: loop-invariant.
  float xnm[8];
#pragma unroll
  for (int j = 0; j < 8; ++j) xnm[j] = xn[rt * 16 + hi * 8 + j];

  // running (act, ~n) best key per C-row; cross-lane reduction deferred.
  unsigned long long best[8];
#pragma unroll
  for (int j = 0; j < 8; ++j) best[j] = 0ull;

  for (int it = 0; it < kN / 32; ++it) {
    const int n = it * 32 + ct * 16 + l16;
    const float* wrow = W + (long)n * kD;

    // B fragments: one clause of 16 global_load_b64, all in flight.
    v2f bfrag[16];
#pragma unroll
    for (int kk = 0; kk < 16; ++kk) {
      const float2 t = *(const float2*)(wrow + kk * 4 + koff);
      bfrag[kk].x = t.x; bfrag[kk].y = t.y;
    }

    // two independent WMMA accumulation chains (break D->C serialization)
    v8f acc0 = {}, acc1 = {};
#pragma unroll
    for (int kk = 0; kk < 16; kk += 2) {
      acc0 = __builtin_amdgcn_wmma_f32_16x16x4_f32(
          false, afrag[kk], false, bfrag[kk], (short)0, acc0, false, false);
      acc1 = __builtin_amdgcn_wmma_f32_16x16x4_f32(
          false, afrag[kk + 1], false, bfrag[kk + 1], (short)0, acc1, false, false);
    }

    const float4 p = np4[n];  // rsum, |w|^2, rsum/64, node_control
    const float base = p.x + 0.001f;
#pragma unroll
    for (int j = 0; j < 8; ++j) {
      float dot = acc0[j] + acc1[j];
      float dist = (xnm[j] + p.y) - 2.0f * dot;
      float dw = dist * p.z;
      dw = (dw != dw) ? 0.0f : dw;  // NaN -> 0
      // act = rsum/(rsum+dw+0.001)*nc ; v_rcp_f32 (~1ulp) is safe: decision
      // margins are ~1e-2 .. 0.67, orders of magnitude above rcp error.
      float act = p.x * __builtin_amdgcn_rcpf(base + dw) * p.w;
      unsigned long long key =
          ((unsigned long long)__float_as_uint(act) << 32) | (unsigned)(~n);
      best[j] = (key > best[j]) ? key : best[j];
    }
  }

  // one cross-lane (16-wide) reduction per row, then ds_max_u64 combine
#pragma unroll
  for (int j = 0; j < 8; ++j) {
    unsigned long long key = best[j];
#pragma unroll
    for (int mask = 1; mask < 16; mask <<= 1) {
      unsigned long long o = __shfl_xor(key, mask, 32);
      key = (o > key) ? o : key;
    }
    if (l16 == 0) atomicMax(&smax[rt * 16 + hi * 8 + j], key);
  }
  __syncthreads();

  // high rows scatter into segment sums (low branch is dead w.r.t. output)
  {
    const int r = tid >> 2, q = tid & 3;
    unsigned long long key = smax[r];
    float amax = __uint_as_float((unsigned)(key >> 32));
    if (amax >= 0.985f) {
      int idx = (int)(~(unsigned)key) & (kN - 1);
      if (q == 0) atomicAdd(&h_cnt[idx], 1.0f);
#pragma unroll
      for (int j = 0; j < 16; ++j) {
        int d = q * 16 + j;
        atomicAdd(&h_sum[(long)idx * kD + d], xs[r * kXsStride + d]);
      }
    }
  }
}

// ---------------------------------------------------------------------------
// kernel 2: per-node epilogue -> the three output slabs
// ---------------------------------------------------------------------------
__global__ __launch_bounds__(64) void som_node_out(
    const float* __restrict__ W, const float* __restrict__ MA,
    const float* __restrict__ h_cnt, const float* __restrict__ h_sum,
    float* __restrict__ out) {
  const int n = blockIdx.x, d = threadIdx.x;
  __shared__ float smx[64], smn[64], ssm[64];
  const long i = (long)n * kD + d;

  const float hc = h_cnt[n];
  const bool hm = hc > 0.0f;
  const float wa = h_sum[i] / fmaxf(hc, 1.0f);
  const float w = W[i];
  // a = LR*DS_BETA = 0.05
  const float nma = 0.05f * fabsf(wa - w) + 0.95f * MA[i];

  smx[d] = nma; smn[d] = nma; ssm[d] = nma;
  __syncthreads();
  for (int s = 32; s > 0; s >>= 1) {
    if (d < s) {
      smx[d] = fmaxf(smx[d], smx[d + s]);
      smn[d] = fminf(smn[d], smn[d + s]);
      ssm[d] += ssm[d + s];
    }
    __syncthreads();
  }
  const float mx = smx[0], mn = smn[0], av = ssm[0] * (1.0f / 64.0f);

  float rel = 1.0f / (1.0f + expf((nma - av) / (mx - mn)));  // EPS_DS = 1
  rel = (rel != rel) ? 1.0f : rel;                            // NaN -> 1

  out[0 * (kN * kD) + i] = hm ? wa : 0.0f;
  out[1 * (kN * kD) + i] = hm ? (w + 0.1f * (wa - w)) : 0.0f;
  out[2 * (kN * kD) + i] = hm ? rel : 0.0f;
}

// ---------------------------------------------------------------------------
extern "C" void kernel_launch(void* const* d_in, const int* in_sizes, int n_in,
                              void* d_out, int out_size, void* d_ws, size_t ws_size,
                              hipStream_t stream) {
  (void)in_sizes; (void)n_in; (void)out_size; (void)ws_size;
  const float* x   = (const float*)d_in[0];   // (B, 64)
  const float* W   = (const float*)d_in[1];   // (1024, 64)
  const float* nc  = (const float*)d_in[2];   // (1024,)
  const float* rel = (const float*)d_in[3];   // (1024, 64)
  const float* ma  = (const float*)d_in[4];   // (1024, 64)
  // d_in[5] = life: provably dead w.r.t. the output, skipped.
  float* out = (float*)d_out;
  float* ws  = (float*)d_ws;

  float4* np4   = (float4*)ws;     // 1024 float4  (4096 floats)
  float*  h_cnt = ws + 4096;       // 1024
  float*  h_sum = ws + 5120;       // 65536  (contiguous with h_cnt)

  som_zero<<<(66560 + 255) / 256, 256, 0, stream>>>(h_cnt, 66560);
  som_nodeparams<<<kN, 64, 0, stream>>>(W, rel, nc, np4);
  som_gemm_assign<<<kB / kRowsPerBlock, 256, 0, stream>>>(x, W, np4, h_cnt, h_sum);
  som_node_out<<<kN, 64, 0, stream>>>(W, ma, h_cnt, h_sum, out);
}